// ConsciousnessMonitor_5007931867609
// MI455X (gfx1250) — compile-verified
//
#include <hip/hip_runtime.h>
#include <math.h>

typedef __attribute__((ext_vector_type(2))) float v2f;
typedef __attribute__((ext_vector_type(8))) float v8f;

namespace {
constexpr int T_ROWS = 32768;   // TIME
constexpr int D_DIM  = 2048;    // STATE_DIM
constexpr int NPART  = 4;
constexpr int NBINS  = 10;
constexpr int SAMPN  = 10;
constexpr int BATCH  = 64;
constexpr int MEMROWS = 100;

// workspace layout (32-bit word offsets)
constexpr int WS_S    = 0;                               // 32768*8 floats: sa[0..3], sb[0..3]
constexpr int WS_MM   = T_ROWS * 8;                      // 16 floats: min[8], max[8]
constexpr int WS_HIST = WS_MM + 16;                      // 4*100 uint joint histograms
constexpr int WS_TVP  = WS_HIST + NPART * NBINS * NBINS; // 8 floats: per-block sum(var)
constexpr int WS_SV2P = WS_TVP + 8;                      // 8 floats: per-block sum(var^2)
constexpr int WS_PD   = WS_SV2P + 8;                     // 45 floats: pairwise distances
}

// ---------------------------------------------------------------- init ------
__global__ void init_kernel(unsigned int* jh) {
  int i = threadIdx.x;
  if (i < NPART * NBINS * NBINS) jh[i] = 0u;
}

// ------------------------------------------------------- WMMA masked GEMM ---
// S[t][p]   = sum_k H[t][k]*mask_p[k] / cnt_p
// S[t][4+p] = (rowsum - masked) / (D - cnt_p)
//
// B weight matrix lives in LDS as floats, 6 padded columns:
//   col 0..3 = masks (0/1), col 4 = ones (rowsum), col 5 = zeros (lanes 5..15)
// Column stride 2052 words => columns are 4 banks apart => conflict-free b64
// reads for lanes 0..4; lanes 5..15 broadcast the same zero-column address.
__global__ __launch_bounds__(256) void gemm_kernel(const float* __restrict__ H,
                                                   const unsigned char* __restrict__ parts,
                                                   float* __restrict__ S) {
  constexpr int WSTRIDE = D_DIM + 4;            // 2052: shifts banks by 4 per column
  __shared__ float wlds[6 * WSTRIDE];
  __shared__ int cnt_s[NPART];
  const int tid = threadIdx.x;
  if (tid < NPART) cnt_s[tid] = 0;
  __syncthreads();
  for (int i = tid; i < 6 * D_DIM; i += 256) {  // uniform trip count (48 iters): no divergence
    const int col = i >> 11;                    // i / 2048
    const int kk  = i & (D_DIM - 1);
    float v;
    if (col < NPART) {
      const unsigned char b = parts[col * D_DIM + kk];
      v = b ? 1.0f : 0.0f;
      if (b) atomicAdd(&cnt_s[col], 1);
    } else {
      v = (col == NPART) ? 1.0f : 0.0f;         // col 4 = ones, col 5 = zeros
    }
    wlds[col * WSTRIDE + kk] = v;
  }
  __syncthreads();

#if __has_builtin(__builtin_amdgcn_wmma_f32_16x16x4_f32)
  // ---- WMMA path: wave-per-16-row tile, f32 16x16x4, K streamed by 4 ----
  const int lane = tid & 31;
  const int wave = tid >> 5;
  const int hf   = lane >> 4;        // lane half: 0 -> K{0,1}, 1 -> K{2,3}
  const int l    = lane & 15;
  const int tileBase = (blockIdx.x * 8 + wave) * 16;
  const float* hrow = H + (size_t)(tileBase + l) * D_DIM;

  const int cidx = (l <= NPART) ? l : (NPART + 1);   // lanes 5..15 -> zero column
  const float* wp = &wlds[cidx * WSTRIDE];

  v8f c = {};
#pragma unroll 8
  for (int k = 0; k < D_DIM; k += 4) {
    const int kk = k + 2 * hf;
    const float2 av = *reinterpret_cast<const float2*>(hrow + kk); // global_load_b64
    const float2 bv = *reinterpret_cast<const float2*>(wp + kk);   // ds_load_b64, branchless
    v2f a; a.x = av.x; a.y = av.y;
    v2f b; b.x = bv.x; b.y = bv.y;
    c = __builtin_amdgcn_wmma_f32_16x16x4_f32(false, a, false, b, (short)0, c, false, false);
  }

  float invA = 1.0f, invB = 1.0f;
  if (l < NPART) {
    const float cn = (float)cnt_s[l];
    invA = 1.0f / cn;
    invB = 1.0f / ((float)D_DIM - cn);
  }
#pragma unroll
  for (int v = 0; v < 8; ++v) {
    const float cv = c[v];                         // C[v + 8*hf][l]
    const float rs = __shfl(cv, 4 + 16 * hf, 32);  // rowsum lives in column-4 lane of same half
    if (l < NPART) {
      const int t = tileBase + 8 * hf + v;
      S[(size_t)t * 8 + l]     = cv * invA;
      S[(size_t)t * 8 + 4 + l] = (rs - cv) * invB;
    }
  }
#else
  // ---- scalar fallback (same launch geometry: 256 blocks x 256 threads) ----
  const int row = blockIdx.x * 256 + tid;
  if (row < T_ROWS) {
    const float* hrow = H + (size_t)row * D_DIM;
    float acc[NPART] = {0.f, 0.f, 0.f, 0.f};
    float rsum = 0.f;
    for (int k = 0; k < D_DIM; ++k) {
      const float h = hrow[k];
      rsum += h;
#pragma unroll
      for (int p = 0; p < NPART; ++p) acc[p] += h * wlds[p * WSTRIDE + k];
    }
#pragma unroll
    for (int p = 0; p < NPART; ++p) {
      const float cn = (float)cnt_s[p];
      S[(size_t)row * 8 + p]     = acc[p] / cn;
      S[(size_t)row * 8 + 4 + p] = (rsum - acc[p]) / ((float)D_DIM - cn);
    }
  }
#endif
}

// ----------------------------------------------------- min/max of 8 series --
__global__ __launch_bounds__(1024) void minmax_kernel(const float* __restrict__ S,
                                                      float* __restrict__ mm) {
  float mn[8], mx[8];
#pragma unroll
  for (int s = 0; s < 8; ++s) { mn[s] = 3.402823e38f; mx[s] = -3.402823e38f; }
  for (int t = threadIdx.x; t < T_ROWS; t += 1024) {
    const float4* p = reinterpret_cast<const float4*>(S + (size_t)t * 8);
    const float4 r0 = p[0], r1 = p[1];
    const float r[8] = {r0.x, r0.y, r0.z, r0.w, r1.x, r1.y, r1.z, r1.w};
#pragma unroll
    for (int s = 0; s < 8; ++s) { mn[s] = fminf(mn[s], r[s]); mx[s] = fmaxf(mx[s], r[s]); }
  }
#pragma unroll
  for (int off = 16; off > 0; off >>= 1) {
#pragma unroll
    for (int s = 0; s < 8; ++s) {
      mn[s] = fminf(mn[s], __shfl_down(mn[s], off, 32));
      mx[s] = fmaxf(mx[s], __shfl_down(mx[s], off, 32));
    }
  }
  __shared__ float smn[32][8], smx[32][8];
  const int lane = threadIdx.x & 31, wid = threadIdx.x >> 5;
  if (lane == 0) {
#pragma unroll
    for (int s = 0; s < 8; ++s) { smn[wid][s] = mn[s]; smx[wid][s] = mx[s]; }
  }
  __syncthreads();
  if (threadIdx.x < 32) {
#pragma unroll
    for (int s = 0; s < 8; ++s) { mn[s] = smn[threadIdx.x][s]; mx[s] = smx[threadIdx.x][s]; }
#pragma unroll
    for (int off = 16; off > 0; off >>= 1) {
#pragma unroll
      for (int s = 0; s < 8; ++s) {
        mn[s] = fminf(mn[s], __shfl_down(mn[s], off, 32));
        mx[s] = fmaxf(mx[s], __shfl_down(mx[s], off, 32));
      }
    }
    if (threadIdx.x == 0) {
#pragma unroll
      for (int s = 0; s < 8; ++s) { mm[s] = mn[s]; mm[8 + s] = mx[s]; }
    }
  }
}

// ------------------------------------------------------ joint histograms ----
__global__ __launch_bounds__(256) void hist_kernel(const float* __restrict__ S,
                                                   const float* __restrict__ mm,
                                                   unsigned int* __restrict__ jh) {
  __shared__ unsigned int h[NPART * NBINS * NBINS];
  for (int i = threadIdx.x; i < NPART * NBINS * NBINS; i += 256) h[i] = 0u;
  __syncthreads();
  float mn[8], mx[8];
#pragma unroll
  for (int s = 0; s < 8; ++s) { mn[s] = mm[s]; mx[s] = mm[8 + s]; }
  for (int t = blockIdx.x * 256 + threadIdx.x; t < T_ROWS; t += gridDim.x * 256) {
    const float4* p = reinterpret_cast<const float4*>(S + (size_t)t * 8);
    const float4 r0 = p[0], r1 = p[1];
    const float r[8] = {r0.x, r0.y, r0.z, r0.w, r1.x, r1.y, r1.z, r1.w};
#pragma unroll
    for (int q = 0; q < NPART; ++q) {
      const float x = (r[q]     - mn[q])     / (mx[q]     - mn[q]     + 1e-6f);
      const float y = (r[4 + q] - mn[4 + q]) / (mx[4 + q] - mn[4 + q] + 1e-6f);
      int xb = (int)(x * (float)NBINS); xb = xb < 0 ? 0 : (xb > NBINS - 1 ? NBINS - 1 : xb);
      int yb = (int)(y * (float)NBINS); yb = yb < 0 ? 0 : (yb > NBINS - 1 ? NBINS - 1 : yb);
      atomicAdd(&h[q * NBINS * NBINS + xb * NBINS + yb], 1u);
    }
  }
  __syncthreads();
  for (int i = threadIdx.x; i < NPART * NBINS * NBINS; i += 256)
    if (h[i]) atomicAdd(&jh[i], h[i]);
}

// ----------------------------------------- per-column variance (ddof=1) -----
__global__ __launch_bounds__(256) void colvar_kernel(const float* __restrict__ state,
                                                     const float* __restrict__ smem,
                                                     float* __restrict__ tvp,
                                                     float* __restrict__ sv2p) {
  const int d = blockIdx.x * 256 + threadIdx.x;   // grid = 8 blocks -> d < 2048 exactly
  float sum = 0.f;
  for (int r = 0; r < MEMROWS; ++r) {
    const float v = (r < BATCH) ? state[(size_t)r * D_DIM + d] : smem[(size_t)r * D_DIM + d];
    sum += v;
  }
  const float mean = sum / (float)MEMROWS;
  float ssd = 0.f;
  for (int r = 0; r < MEMROWS; ++r) {
    const float v = (r < BATCH) ? state[(size_t)r * D_DIM + d] : smem[(size_t)r * D_DIM + d];
    const float df = v - mean;
    ssd += df * df;
  }
  const float var = ssd / (float)(MEMROWS - 1);
  float s1 = var, s2 = var * var;
#pragma unroll
  for (int off = 16; off > 0; off >>= 1) {
    s1 += __shfl_down(s1, off, 32);
    s2 += __shfl_down(s2, off, 32);
  }
  __shared__ float a1[8], a2[8];
  const int lane = threadIdx.x & 31, wid = threadIdx.x >> 5;
  if (lane == 0) { a1[wid] = s1; a2[wid] = s2; }
  __syncthreads();
  if (threadIdx.x == 0) {
    float t1 = 0.f, t2 = 0.f;
    for (int w = 0; w < 8; ++w) { t1 += a1[w]; t2 += a2[w]; }
    tvp[blockIdx.x] = t1;
    sv2p[blockIdx.x] = t2;
  }
}

// ------------------------------------------- pairwise sample distances ------
__global__ __launch_bounds__(256) void pair_kernel(const float* __restrict__ state,
                                                   const float* __restrict__ smem,
                                                   const int* __restrict__ sidx,
                                                   float* __restrict__ pd) {
  // decode pair index -> (i, j), i < j, over SAMPN=10 rows (45 pairs)
  int i = 0, rem = blockIdx.x;
  while (rem >= (SAMPN - 1 - i)) { rem -= (SAMPN - 1 - i); ++i; }
  const int j = i + 1 + rem;
  const int ri = sidx[i], rj = sidx[j];
  const float* a = (ri < BATCH) ? state + (size_t)ri * D_DIM : smem + (size_t)ri * D_DIM;
  const float* b = (rj < BATCH) ? state + (size_t)rj * D_DIM : smem + (size_t)rj * D_DIM;
  float s = 0.f;
  for (int d = threadIdx.x; d < D_DIM; d += 256) {
    const float df = a[d] - b[d];
    s += df * df;
  }
#pragma unroll
  for (int off = 16; off > 0; off >>= 1) s += __shfl_down(s, off, 32);
  __shared__ float red[8];
  const int lane = threadIdx.x & 31, wid = threadIdx.x >> 5;
  if (lane == 0) red[wid] = s;
  __syncthreads();
  if (threadIdx.x == 0) {
    float tot = 0.f;
    for (int w = 0; w < 8; ++w) tot += red[w];
    pd[blockIdx.x] = sqrtf(fmaxf(tot, 0.f));
  }
}

// -------------------------------------------------------------- finalize ----
__global__ void finalize_kernel(const unsigned int* __restrict__ jh,
                                const float* __restrict__ tvp,
                                const float* __restrict__ sv2p,
                                const float* __restrict__ pd,
                                float* __restrict__ out) {
  if (threadIdx.x != 0 || blockIdx.x != 0) return;
  float tv = 0.f, sv2 = 0.f;
  for (int i = 0; i < 8; ++i) { tv += tvp[i]; sv2 += sv2p[i]; }
  const float tvs = (tv > 0.f) ? tv : 1.0f;
  const float sumnv2 = sv2 / (tvs * tvs);
  const float eff = (tv > 0.f) ? (1.0f / (sumnv2 + 1e-6f)) : 1.0f;
  float dsum = 0.f;
  for (int i = 0; i < 45; ++i) dsum += pd[i];                 // one triangle
  const float avg = (2.0f * dsum) / ((float)(SAMPN * (SAMPN - 1)) + 1e-6f);
  const float diff = sqrtf(tv) * avg;

  float mi[NPART];
  for (int p = 0; p < NPART; ++p) {
    const unsigned int* hp = jh + p * NBINS * NBINS;
    float tot = 0.f;
    for (int i = 0; i < NBINS * NBINS; ++i) tot += (float)hp[i];
    const float inv = 1.0f / (tot + 1e-10f);
    float px[NBINS], py[NBINS];
    for (int i = 0; i < NBINS; ++i) { px[i] = 0.f; py[i] = 0.f; }
    for (int x = 0; x < NBINS; ++x)
      for (int y = 0; y < NBINS; ++y) {
        const float jv = (float)hp[x * NBINS + y] * inv;
        px[x] += jv; py[y] += jv;
      }
    float m = 0.f;
    for (int x = 0; x < NBINS; ++x)
      for (int y = 0; y < NBINS; ++y) {
        const float jv = (float)hp[x * NBINS + y] * inv;
        m += jv * logf((jv + 1e-10f) / (px[x] * py[y] + 1e-10f));
      }
    mi[p] = fmaxf(m, 0.f);
  }
  const float integ = fminf(fminf(mi[0], mi[1]), fminf(mi[2], mi[3]));
  out[0] = integ + tanhf(diff);
  out[1] = diff;
  out[2] = eff;
  out[3] = tv;
  out[4] = integ;
  for (int p = 0; p < NPART; ++p) out[5 + p] = mi[p];
}

// ---------------------------------------------------------------------------
extern "C" void kernel_launch(void* const* d_in, const int* in_sizes, int n_in,
                              void* d_out, int out_size, void* d_ws, size_t ws_size,
                              hipStream_t stream) {
  const float* state         = (const float*)d_in[0];         // (64, 2048) f32
  const float* state_memory  = (const float*)d_in[1];         // (100, 2048) f32
  const float* state_history = (const float*)d_in[2];         // (32768, 2048) f32
  const unsigned char* parts = (const unsigned char*)d_in[3]; // (4, 2048) bool
  const int* sample_idx      = (const int*)d_in[4];           // (10,) i32
  float* out = (float*)d_out;                                 // 9 floats

  float* ws   = (float*)d_ws;
  float* S    = ws + WS_S;
  float* mm   = ws + WS_MM;
  unsigned int* jh = (unsigned int*)(ws + WS_HIST);
  float* tvp  = ws + WS_TVP;
  float* sv2p = ws + WS_SV2P;
  float* pd   = ws + WS_PD;

  hipLaunchKernelGGL(init_kernel,     dim3(1),   dim3(512),  0, stream, jh);
  hipLaunchKernelGGL(gemm_kernel,     dim3(256), dim3(256),  0, stream, state_history, parts, S);
  hipLaunchKernelGGL(minmax_kernel,   dim3(1),   dim3(1024), 0, stream, S, mm);
  hipLaunchKernelGGL(hist_kernel,     dim3(64),  dim3(256),  0, stream, S, mm, jh);
  hipLaunchKernelGGL(colvar_kernel,   dim3(8),   dim3(256),  0, stream, state, state_memory, tvp, sv2p);
  hipLaunchKernelGGL(pair_kernel,     dim3(45),  dim3(256),  0, stream, state, state_memory, sample_idx, pd);
  hipLaunchKernelGGL(finalize_kernel, dim3(1),   dim3(32),   0, stream, jh, tvp, sv2p, pd, out);
}